// PolylineSubGraph_61555471286758
// MI455X (gfx1250) — compile-verified
//
#include <hip/hip_runtime.h>

// PolylineSubGraph fused kernel for gfx1250 (MI455X), fp32 via V_WMMA_F32_16X16X4_F32.
// One workgroup per (batch, polyline); whole 3-layer pipeline in LDS.

typedef float v2f __attribute__((ext_vector_type(2)));
typedef float v8f __attribute__((ext_vector_type(8)));

#define BB   4      // batch
#define PP   64     // polylines
#define NN   48     // vectors per polyline (3 x 16 WMMA row tiles)
#define FF   32     // base features
#define LDS_STRIDE 260   // row stride (floats); != multiple of 64 -> conflict-free ds loads
#define NWAVES 8
#define NTHREADS 256

struct PolyParams {
    const float* w[24];   // per layer: we, be, wE, bE, w1, b1, w2, b2
};

// C(lds)[48 x Nout] = act( A(lds)[48 x K] @ Wg[K x Nout] + bias )
// A rows at LDS_STRIDE. Wg row-major with row stride == Nout. All 8 waves participate.
__device__ __forceinline__ void block_gemm(const float* Al, float* Cl,
                                           const float* __restrict__ Wg,
                                           const float* __restrict__ bias,
                                           int K, int Nout, bool relu)
{
    const int wid   = threadIdx.x >> 5;
    const int lane  = threadIdx.x & 31;
    const int m16   = lane & 15;
    const int kh    = (lane >> 4) << 1;   // 0 or 2: which K-pair this lane-half holds
    const int rbase = (lane >> 4) << 3;   // 0 or 8: C/D row offset for this lane-half
    const int strips = Nout >> 4;

    for (int s = wid; s < strips; s += NWAVES) {
        const int col = (s << 4) + m16;
        v8f acc0 = {}, acc1 = {}, acc2 = {};
        for (int k0 = 0; k0 < K; k0 += 4) {
            const int kk = k0 + kh;
            // B fragment (4x16 f32): VGPR0 holds rows {0,2}, VGPR1 rows {1,3} by lane-half
            v2f bf;
            bf.x = Wg[(kk + 0) * Nout + col];
            bf.y = Wg[(kk + 1) * Nout + col];
            // A fragments (16x4 f32): lane m16 = row, lane-half selects K pair
            v2f a0 = *(const v2f*)&Al[( 0 + m16) * LDS_STRIDE + kk];
            v2f a1 = *(const v2f*)&Al[(16 + m16) * LDS_STRIDE + kk];
            v2f a2 = *(const v2f*)&Al[(32 + m16) * LDS_STRIDE + kk];
            acc0 = __builtin_amdgcn_wmma_f32_16x16x4_f32(false, a0, false, bf, (short)0, acc0, false, false);
            acc1 = __builtin_amdgcn_wmma_f32_16x16x4_f32(false, a1, false, bf, (short)0, acc1, false, false);
            acc2 = __builtin_amdgcn_wmma_f32_16x16x4_f32(false, a2, false, bf, (short)0, acc2, false, false);
        }
        const float bv = bias ? bias[col] : 0.0f;
        #pragma unroll
        for (int r = 0; r < 8; ++r) {
            float c0 = acc0[r] + bv, c1 = acc1[r] + bv, c2 = acc2[r] + bv;
            if (relu) { c0 = fmaxf(c0, 0.f); c1 = fmaxf(c1, 0.f); c2 = fmaxf(c2, 0.f); }
            Cl[( 0 + rbase + r) * LDS_STRIDE + col] = c0;
            Cl[(16 + rbase + r) * LDS_STRIDE + col] = c1;
            Cl[(32 + rbase + r) * LDS_STRIDE + col] = c2;
        }
    }
}

__global__ __launch_bounds__(NTHREADS)
void polyline_subgraph_kernel(const float* __restrict__ x,
                              const unsigned char* __restrict__ masks,
                              PolyParams pw,
                              float* __restrict__ out)
{
    // Two ping-pong activation buffers: ~100 KB of the 320 KB/WGP LDS.
    __shared__ float Pb[NN * LDS_STRIDE];
    __shared__ float Qb[NN * LDS_STRIDE];
    __shared__ float Sb[256];     // S_b = (sum_j m_j H_j) @ wE_bot   (2d <= 256)
    __shared__ float hsum[128];   // sum_j m_j H_j                     (d  <= 128)
    __shared__ float mval[NN];
    __shared__ float s_cnt, s_inv;

    const int bp = blockIdx.x;                 // polyline id in [0, B*P)
    const float* xg = x + (size_t)bp * NN * FF;

    if (threadIdx.x < NN)
        mval[threadIdx.x] = masks[(size_t)bp * NN + threadIdx.x] ? 1.0f : 0.0f;
    __syncthreads();
    if (threadIdx.x == 0) {
        float c = 0.f;
        #pragma unroll
        for (int j = 0; j < NN; ++j) c += mval[j];
        s_cnt = c;
        s_inv = 1.0f / (c + 1e-6f);
    }
    // stage input X -> Pb
    for (int t = threadIdx.x; t < NN * FF; t += NTHREADS) {
        int r = t / FF, c = t % FF;
        Pb[r * LDS_STRIDE + c] = xg[t];
    }
    __syncthreads();

    int d = FF;
    #pragma unroll
    for (int layer = 0; layer < 3; ++layer) {
        const float* we = pw.w[layer * 8 + 0];
        const float* be = pw.w[layer * 8 + 1];
        const float* wE = pw.w[layer * 8 + 2];
        const float* bE = pw.w[layer * 8 + 3];
        const float* w1 = pw.w[layer * 8 + 4];
        const float* b1 = pw.w[layer * 8 + 5];
        const float* w2 = pw.w[layer * 8 + 6];
        const float* b2 = pw.w[layer * 8 + 7];
        const int d2 = 2 * d;

        // H = relu(X @ we + be) : Pb -> Qb
        block_gemm(Pb, Qb, we, be, d, d, true);
        __syncthreads();

        // A = H @ wE[:d] + bE : Qb -> Pb   (edge i-part)
        block_gemm(Qb, Pb, wE, bE, d, d2, false);
        // hsum[k] = sum_j m_j * H[j][k]   (reads Qb, independent of A writes)
        if (threadIdx.x < d) {
            float s = 0.f;
            for (int j = 0; j < NN; ++j) s += mval[j] * Qb[j * LDS_STRIDE + threadIdx.x];
            hsum[threadIdx.x] = s;
        }
        __syncthreads();

        // S_b[c] = hsum @ wE[d:, c]   (rank-1 collapse of the masked edge sum)
        if (threadIdx.x < d2) {
            float s = 0.f;
            const float* wb = wE + (size_t)d * d2;
            for (int k = 0; k < d; ++k) s += hsum[k] * wb[k * d2 + threadIdx.x];
            Sb[threadIdx.x] = s;
        }
        __syncthreads();

        // node_i = (cnt * A_i + S_b) / (cnt + eps), in place in Pb
        const float cnt = s_cnt, inv = s_inv;
        for (int t = threadIdx.x; t < NN * d2; t += NTHREADS) {
            int r = t / d2, c = t % d2;
            float* p = &Pb[r * LDS_STRIDE + c];
            *p = (cnt * (*p) + Sb[c]) * inv;
        }
        __syncthreads();

        // T = relu(node @ w1 + b1) : Pb -> Qb
        block_gemm(Pb, Qb, w1, b1, d2, d2, true);
        __syncthreads();

        // X_next = T @ w2 + b2 : Qb -> Pb
        block_gemm(Qb, Pb, w2, b2, d2, d2, false);
        __syncthreads();

        d = d2;
    }

    // masked mean pool over nodes: out[bp][c] = inv * sum_i m_i * X[i][c]
    for (int c = threadIdx.x; c < d; c += NTHREADS) {
        float s = 0.f;
        #pragma unroll 4
        for (int i = 0; i < NN; ++i) s += mval[i] * Pb[i * LDS_STRIDE + c];
        out[(size_t)bp * d + c] = s * s_inv;
    }
}

extern "C" void kernel_launch(void* const* d_in, const int* in_sizes, int n_in,
                              void* d_out, int out_size, void* d_ws, size_t ws_size,
                              hipStream_t stream)
{
    (void)in_sizes; (void)n_in; (void)d_ws; (void)ws_size; (void)out_size;
    const float* x = (const float*)d_in[0];
    const unsigned char* masks = (const unsigned char*)d_in[1];
    PolyParams pw;
    for (int i = 0; i < 24; ++i) pw.w[i] = (const float*)d_in[2 + i];
    float* out = (float*)d_out;

    dim3 grid(BB * PP);   // 256 independent polylines
    dim3 block(NTHREADS); // 8 waves of 32
    polyline_subgraph_kernel<<<grid, block, 0, stream>>>(x, masks, pw, out);
}